// TopK_48498770707332
// MI455X (gfx1250) — compile-verified
//
#include <hip/hip_runtime.h>
#include <stdint.h>

#define DICT 24576
#define KSEL 64
#define THREADS 256
#define GROUPS_PER_THREAD (DICT / (THREADS * 4))  // 24 float4 groups per thread
#define TIE_CAP 256
#define BOUND 2.5f   // prune guard: used only if >=KSEL elements exceed it (then exact)

// The async-to-LDS builtins take typed vector pointers:
//   param0: int __vector(4) addrspace(1)*   (global side)
//   param1: int __vector(4) addrspace(3)*   (LDS side)
typedef int v4i __attribute__((vector_size(16)));
typedef __attribute__((address_space(1))) v4i* gptr_v4i;
typedef __attribute__((address_space(3))) v4i* lptr_v4i;

// Casts go via integer to avoid generic->AS pointer cast restrictions.
// For LDS, the low 32 bits of the flat address are the LDS byte offset
// (flat->LDS mapping discards upper bits per CDNA5 ISA 4.3.4).
#define AS1P(p) ((gptr_v4i)(uintptr_t)(p))
#define AS3P(p) ((lptr_v4i)(uint32_t)(uintptr_t)(p))

__device__ __forceinline__ unsigned int fkey(float f) {
  unsigned int u = __float_as_uint(f);
  // Monotonic: larger float -> larger unsigned key
  return (u & 0x80000000u) ? ~u : (u | 0x80000000u);
}

__device__ __forceinline__ void async_load16(const float* g, float* l) {
#if __has_builtin(__builtin_amdgcn_global_load_async_to_lds_b128)
  __builtin_amdgcn_global_load_async_to_lds_b128(AS1P(g), AS3P(l), 0, 0);
#else
  asm volatile("global_load_async_to_lds_b128 %0, %1, off"
               :: "v"((uint32_t)(uintptr_t)(l)), "v"(g)
               : "memory");
#endif
}

__device__ __forceinline__ void async_store16(float* g, const float* l) {
#if __has_builtin(__builtin_amdgcn_global_store_async_from_lds_b128)
  __builtin_amdgcn_global_store_async_from_lds_b128(AS1P(g), AS3P(l), 0, 0);
#else
  asm volatile("global_store_async_from_lds_b128 %0, %1, off"
               :: "v"(g), "v"((uint32_t)(uintptr_t)(l))
               : "memory");
#endif
}

__device__ __forceinline__ void wait_async0() {
#if __has_builtin(__builtin_amdgcn_s_wait_asynccnt)
  __builtin_amdgcn_s_wait_asynccnt(0);
#else
  asm volatile("s_wait_asynccnt 0" ::: "memory");
#endif
}

__global__ __launch_bounds__(THREADS)
void topk_scatter_kernel(const float* __restrict__ z, float* __restrict__ out) {
  // One workgroup owns one full row (96 KB) staged in LDS (320 KB/WGP).
  __shared__ __align__(16) float row[DICT];
  __shared__ unsigned int hist[256];
  __shared__ unsigned int tieIdx[TIE_CAP];
  __shared__ unsigned int keepIdx[KSEL];
  __shared__ unsigned int ctrl[4];  // [0]=prefix, [1]=remaining, [2]=tie count, [3]=above-bound count

  const int t = threadIdx.x;
  const size_t base = (size_t)blockIdx.x * (size_t)DICT;
  const float* src = z + base;
  float* dst = out + base;

  // ---- Stage row into LDS with async DMA (ASYNCcnt path) ----
  for (int j = 0; j < GROUPS_PER_THREAD; ++j) {
    const int g = (t + THREADS * j) * 4;
    async_load16(src + g, &row[g]);
  }
  if (t == 0) { ctrl[0] = 0u; ctrl[1] = (unsigned int)KSEL; ctrl[2] = 0u; ctrl[3] = 0u; }
  wait_async0();        // each wave waits for its own async loads
  __syncthreads();      // then all waves' LDS writes are visible

  // ---- Prune guard: count elements strictly above BOUND (no per-element atomics) ----
  const unsigned int keyB = fkey(BOUND);
  {
    unsigned int myc = 0u;
    for (int j = 0; j < GROUPS_PER_THREAD; ++j) {
      const int g = t + THREADS * j;
      const float4 v = reinterpret_cast<const float4*>(row)[g];
      myc += (fkey(v.x) > keyB) ? 1u : 0u;
      myc += (fkey(v.y) > keyB) ? 1u : 0u;
      myc += (fkey(v.z) > keyB) ? 1u : 0u;
      myc += (fkey(v.w) > keyB) ? 1u : 0u;
    }
    if (myc) atomicAdd(&ctrl[3], myc);
  }
  __syncthreads();
  // If >=KSEL elements exceed BOUND, the k-th largest is > BOUND, so the top-k is a
  // subset of {key > keyB}: restricting the radix select to that set is exact and
  // eliminates LDS-atomic contention on the dense low bins. Otherwise, no pruning.
  const unsigned int keyLim = (ctrl[3] >= (unsigned int)KSEL) ? keyB : 0u;

  // ---- Exact radix select: 4 rounds x 8 bits, MSB first (over candidate set) ----
  for (int r = 0; r < 4; ++r) {
    const int shift = 24 - 8 * r;
    hist[t] = 0u;
    __syncthreads();
    const unsigned int pfx = ctrl[0];
    const unsigned int pmask = (r == 0) ? 0u : (0xFFFFFFFFu << (shift + 8));
    for (int j = 0; j < GROUPS_PER_THREAD; ++j) {
      const int g = t + THREADS * j;  // conflict-free strided float4 reads
      const float4 v = reinterpret_cast<const float4*>(row)[g];
      const float vv[4] = {v.x, v.y, v.z, v.w};
#pragma unroll
      for (int c = 0; c < 4; ++c) {
        const unsigned int k = fkey(vv[c]);
        if (k > keyLim && (k & pmask) == pfx) atomicAdd(&hist[(k >> shift) & 0xFFu], 1u);
      }
    }
    __syncthreads();
    if (t == 0) {
      const unsigned int rem = ctrl[1];
      unsigned int cum = 0u;
      int b = 255;
      for (; b > 0; --b) {
        const unsigned int c = hist[b];
        if (cum + c >= rem) break;
        cum += c;
      }
      ctrl[0] = pfx | ((unsigned int)b << shift);
      ctrl[1] = rem - cum;  // still needed among keys with this byte
    }
    __syncthreads();
  }

  const unsigned int T   = ctrl[0];  // exact 64th-largest key (T > keyLim by construction)
  const unsigned int rem = ctrl[1];  // how many elements == T to keep (lowest index first)

  // ---- Gather indices of exact ties (usually rem==1, c_eq==1) ----
  for (int j = 0; j < GROUPS_PER_THREAD; ++j) {
    const int g = t + THREADS * j;
    const float4 v = reinterpret_cast<const float4*>(row)[g];
    const float vv[4] = {v.x, v.y, v.z, v.w};
#pragma unroll
    for (int c = 0; c < 4; ++c) {
      if (fkey(vv[c]) == T) {
        const unsigned int p = atomicAdd(&ctrl[2], 1u);
        if (p < TIE_CAP) tieIdx[p] = (unsigned int)(4 * g + c);
      }
    }
  }
  __syncthreads();
  unsigned int ecnt = ctrl[2];
  if (ecnt > TIE_CAP) ecnt = TIE_CAP;
  if ((unsigned int)t < ecnt) {
    const unsigned int my = tieIdx[t];
    unsigned int rank = 0u;
    for (unsigned int e = 0; e < ecnt; ++e) rank += (tieIdx[e] < my) ? 1u : 0u;
    if (rank < rem) keepIdx[rank] = my;  // keep the `rem` lowest indices
  }
  __syncthreads();

  // ---- Zero non-selected elements in place in LDS ----
  for (int j = 0; j < GROUPS_PER_THREAD; ++j) {
    const int g = t + THREADS * j;
    float4 v = reinterpret_cast<float4*>(row)[g];
    float vv[4] = {v.x, v.y, v.z, v.w};
#pragma unroll
    for (int c = 0; c < 4; ++c) {
      const unsigned int k = fkey(vv[c]);
      bool keep = (k > T);
      if (k == T) {
        const unsigned int idx = (unsigned int)(4 * g + c);
        for (unsigned int e = 0; e < rem; ++e) keep = keep || (keepIdx[e] == idx);
      }
      if (!keep) vv[c] = 0.0f;
    }
    v.x = vv[0]; v.y = vv[1]; v.z = vv[2]; v.w = vv[3];
    reinterpret_cast<float4*>(row)[g] = v;
  }
  __syncthreads();

  // ---- Stream filtered row back with async DMA ----
  for (int j = 0; j < GROUPS_PER_THREAD; ++j) {
    const int g = (t + THREADS * j) * 4;
    async_store16(dst + g, &row[g]);
  }
  wait_async0();
}

extern "C" void kernel_launch(void* const* d_in, const int* in_sizes, int n_in,
                              void* d_out, int out_size, void* d_ws, size_t ws_size,
                              hipStream_t stream) {
  const float* z = (const float*)d_in[0];
  float* out = (float*)d_out;
  // in_sizes[0] = BATCH * DICT; k is fixed at 64 per the reference.
  const int batch = in_sizes[0] / DICT;
  topk_scatter_kernel<<<batch, THREADS, 0, stream>>>(z, out);
}